// WeightedAttnReadout_27762668601744
// MI455X (gfx1250) — compile-verified
//
#include <hip/hip_runtime.h>
#include <hip/hip_bf16.h>
#include <stdint.h>

#define N_  262144
#define D_  256
#define B_  1024
#define H_  8
#define HD_ 32
#define EPS_ 1e-5f
#define MEAN_RES_SCALE_ 0.2f
#define INV_SQRT_HD_ 0.17677669529663687f   // 1/sqrt(32)
#define NEG_BIG_ (-3.4e38f)

typedef __attribute__((ext_vector_type(2))) float v2f;
typedef __attribute__((ext_vector_type(4))) float v4f;
typedef __attribute__((ext_vector_type(8))) float v8f;

__device__ __forceinline__ float wred(float v) {
#pragma unroll
  for (int off = 16; off > 0; off >>= 1) v += __shfl_xor(v, off, 32);
  return v;
}
__device__ __forceinline__ v4f splat4(float x) { v4f r = {x, x, x, x}; return r; }
__device__ __forceinline__ v4f vmax4(v4f a, v4f b) {
  v4f r; r.x = fmaxf(a.x, b.x); r.y = fmaxf(a.y, b.y);
  r.z = fmaxf(a.z, b.z); r.w = fmaxf(a.w, b.w); return r;
}
__device__ __forceinline__ float hsum4(v4f a) { return (a.x + a.y) + (a.z + a.w); }
__device__ __forceinline__ v4f exp4(v4f a) {
  v4f r; r.x = expf(a.x); r.y = expf(a.y); r.z = expf(a.z); r.w = expf(a.w); return r;
}

// ---------------------------------------------------------------- K0: segment offsets
__global__ void seg_offsets_k(const int* __restrict__ batch, int* __restrict__ seg) {
  int i = blockIdx.x * blockDim.x + threadIdx.x;
  if (i >= N_) return;
  int b = batch[i];
  if (i == 0) {
    for (int g = 0; g <= b; ++g) seg[g] = 0;
  } else {
    int pb = batch[i - 1];
    for (int g = pb + 1; g <= b; ++g) seg[g] = i;
  }
  if (i == N_ - 1) {
    for (int g = b + 1; g <= B_; ++g) seg[g] = N_;
  }
}

// ---------------------------------------------------------------- K1: per-graph mean/max pooling (b128 streams)
// 256 threads = 4 node-subgroups x 64 channel-quads
__global__ void pool_k(const float* __restrict__ h, const int* __restrict__ seg,
                       float* __restrict__ meanr, float* __restrict__ maxr) {
  __shared__ v4f rs[256];
  __shared__ v4f rm[256];
  int b = blockIdx.x, t = threadIdx.x;
  int g = t >> 6, c4 = t & 63;
  int s0 = seg[b], s1 = seg[b + 1];
  const v4f* h4 = (const v4f*)h;            // 64 v4f per node row
  v4f sum = splat4(0.f), mx = splat4(NEG_BIG_);
  for (int n = s0 + g; n < s1; n += 4) {
    v4f x = h4[(size_t)n * 64 + c4];
    sum += x;
    mx = vmax4(mx, x);
  }
  rs[t] = sum; rm[t] = mx;
  __syncthreads();
  if (g == 0) {
    v4f S = ((rs[t] + rs[t + 64]) + (rs[t + 128] + rs[t + 192]));
    v4f M = vmax4(vmax4(rm[t], rm[t + 64]), vmax4(rm[t + 128], rm[t + 192]));
    float inv = 1.f / (float)(s1 - s0);
    ((v4f*)meanr)[(size_t)b * 64 + c4] = S * inv;
    ((v4f*)maxr)[(size_t)b * 64 + c4]  = M;
  }
}

// ---------------------------------------------------------------- K2a: q_pre = [mean|max] @ Wq^T  (f32 WMMA)
__global__ void q_gemm_k(const float* __restrict__ meanr, const float* __restrict__ maxr,
                         const float* __restrict__ Wq, float* __restrict__ qpre) {
  int wave = (blockIdx.x * blockDim.x + threadIdx.x) >> 5;  // 1024 tiles
  int lane = threadIdx.x & 31;
  int mt = wave >> 4;          // 64 row tiles of B_
  int nt = wave & 15;          // 16 col tiles of D_
  int r  = lane & 15;
  int kh = (lane >> 4) << 1;   // 0 or 2 : K sub-offset per ISA f32 A/B layout
  v8f acc = {};
  int arow = (mt * 16 + r) * D_;
  int brow = (nt * 16 + r) * (2 * D_);
#pragma unroll 4
  for (int kk = 0; kk < 2 * D_; kk += 4) {
    int k0 = kk + kh;
    v2f a, bm;
    if (k0 < D_) { a.x = meanr[arow + k0]; a.y = meanr[arow + k0 + 1]; }
    else         { a.x = maxr[arow + k0 - D_]; a.y = maxr[arow + k0 - D_ + 1]; }
    bm.x = Wq[brow + k0]; bm.y = Wq[brow + k0 + 1];
    acc = __builtin_amdgcn_wmma_f32_16x16x4_f32(false, a, false, bm, (short)0, acc,
                                                false, false);
  }
  int mb  = mt * 16 + ((lane >> 4) << 3);
  int col = nt * 16 + r;
#pragma unroll
  for (int i = 0; i < 8; ++i) qpre[(mb + i) * D_ + col] = acc[i];
}

// ---------------------------------------------------------------- K2b: LayerNorm of q
__global__ void q_ln_k(const float* __restrict__ qpre, const float* __restrict__ w,
                       const float* __restrict__ bi, float* __restrict__ q) {
  __shared__ float red[D_];
  int b = blockIdx.x, t = threadIdx.x;
  float x = qpre[b * D_ + t];
  red[t] = x; __syncthreads();
  for (int st = 128; st > 0; st >>= 1) { if (t < st) red[t] += red[t + st]; __syncthreads(); }
  float mu = red[0] * (1.f / D_);
  __syncthreads();
  float d = x - mu;
  red[t] = d * d; __syncthreads();
  for (int st = 128; st > 0; st >>= 1) { if (t < st) red[t] += red[t + st]; __syncthreads(); }
  float var = red[0] * (1.f / D_);
  q[b * D_ + t] = d * rsqrtf(var + EPS_) * w[t] + bi[t];
}

// ---------------------------------------------------------------- K2c: P[b,h,:] = sum_j q[b,h*32+j] * Wk[h*32+j,:]
__global__ void p_compute_k(const float* __restrict__ q, const float* __restrict__ Wk,
                            float* __restrict__ P) {
  int bh = blockIdx.x;            // B_*H_ blocks
  int b = bh >> 3, hh = bh & 7;
  int c = threadIdx.x;
  const float* qrow = q + b * D_ + hh * HD_;
  float acc = 0.f;
#pragma unroll 4
  for (int j = 0; j < HD_; ++j) acc += qrow[j] * Wk[(hh * HD_ + j) * D_ + c];
  P[(size_t)bh * D_ + c] = acc;
}

// ---------------------------------------------------------------- K3: per-node LN stats + logits (1 wave/node, b128)
__global__ void logits_k(const float* __restrict__ h, const int* __restrict__ batch,
                         const uint8_t* __restrict__ cdr, const uint8_t* __restrict__ ifc,
                         const float* __restrict__ lnw, const float* __restrict__ lnb,
                         const float* __restrict__ P, const float* __restrict__ cdr_b,
                         const float* __restrict__ if_b, const float* __restrict__ lscale,
                         float* __restrict__ logit, float* __restrict__ mu_n,
                         float* __restrict__ rstd_n) {
  int wave = threadIdx.x >> 5;
  int lane = threadIdx.x & 31;
  int n = blockIdx.x * 8 + wave;
  const v4f* row4 = (const v4f*)(h + (size_t)n * D_);
  v4f x0 = row4[lane], x1 = row4[lane + 32];
  float sum = hsum4(x0) + hsum4(x1);
  sum = wred(sum);
  float mu = sum * (1.f / D_);
  v4f d0 = x0 - mu, d1 = x1 - mu;
  float vs = hsum4(d0 * d0) + hsum4(d1 * d1);
  vs = wred(vs);
  float rstd = rsqrtf(vs * (1.f / D_) + EPS_);
  const v4f* w4 = (const v4f*)lnw;
  const v4f* b4 = (const v4f*)lnb;
  v4f hl0 = d0 * rstd * w4[lane] + b4[lane];
  v4f hl1 = d1 * rstd * w4[lane + 32] + b4[lane + 32];
  int bg = batch[n];
  const float* Pb = P + (size_t)bg * H_ * D_;
  float scale = lscale[0] * INV_SQRT_HD_;
  float mb = cdr_b[0] * (cdr[n] ? 1.f : 0.f) + if_b[0] * (ifc[n] ? 1.f : 0.f);
  float lg[8];
#pragma unroll
  for (int hh = 0; hh < H_; ++hh) {
    const v4f* P4 = (const v4f*)(Pb + hh * D_);
    float p = hsum4(hl0 * P4[lane]) + hsum4(hl1 * P4[lane + 32]);
    p = wred(p);
    lg[hh] = p * scale + mb;
  }
  if (lane == 0) {
    v4f o0 = {lg[0], lg[1], lg[2], lg[3]};
    v4f o1 = {lg[4], lg[5], lg[6], lg[7]};
    ((v4f*)logit)[(size_t)n * 2]     = o0;
    ((v4f*)logit)[(size_t)n * 2 + 1] = o1;
    mu_n[n] = mu;
    rstd_n[n] = rstd;
  }
}

// ---------------------------------------------------------------- K4: segment softmax stats (max, 1/sum), b128 logit stream
__global__ void stats_k(const float* __restrict__ logit, const int* __restrict__ seg,
                        float* __restrict__ m, float* __restrict__ zinv) {
  __shared__ v4f red[256];
  int b = blockIdx.x, t = threadIdx.x;
  int s0 = seg[b], s1 = seg[b + 1];
  const v4f* l4 = (const v4f*)logit;         // quad i covers heads (i&1)*4..+3 of node i>>1
  long beg = (long)s0 * 2, end = (long)s1 * 2;
  v4f pm = splat4(NEG_BIG_);
  for (long i = beg + t; i < end; i += 256) pm = vmax4(pm, l4[i]);
  red[t] = pm; __syncthreads();
  for (int st = 128; st >= 2; st >>= 1) {      // parity-preserving: head group = t&1
    if (t < st) red[t] = vmax4(red[t], red[t + st]);
    __syncthreads();
  }
  if (t < H_) m[b * H_ + t] = ((const float*)red)[t];   // red[0]=heads0-3, red[1]=heads4-7
  v4f mh = red[t & 1];
  __syncthreads();
  v4f ps = splat4(0.f);
  for (long i = beg + t; i < end; i += 256) ps += exp4(l4[i] - mh);
  red[t] = ps; __syncthreads();
  for (int st = 128; st >= 2; st >>= 1) {
    if (t < st) red[t] += red[t + st];
    __syncthreads();
  }
  if (t < H_) zinv[b * H_ + t] = 1.f / ((const float*)red)[t];
}

// ---------------------------------------------------------------- K5: s[b,h,:] = sum_n attn[n,h]*h_ln[n,:]  (b128)
// 256 threads = 4 node-subgroups x 64 channel-quads
__global__ void s_accum_k(const float* __restrict__ h, const int* __restrict__ seg,
                          const float* __restrict__ logit, const float* __restrict__ m,
                          const float* __restrict__ zinv, const float* __restrict__ mu_n,
                          const float* __restrict__ rstd_n, const float* __restrict__ lnw,
                          const float* __restrict__ lnb, float* __restrict__ s) {
  __shared__ v4f attn4[128];                 // 64 nodes x 8 heads
  __shared__ v4f red[256];
  const float* attnf = (const float*)attn4;
  int b = blockIdx.x, t = threadIdx.x;
  int g = t >> 6, c4 = t & 63;
  int s0 = seg[b], s1 = seg[b + 1];
  const v4f* h4 = (const v4f*)h;
  v4f w = ((const v4f*)lnw)[c4];
  v4f bi = ((const v4f*)lnb)[c4];
  v4f mh0 = ((const v4f*)(m + b * H_))[0], mh1 = ((const v4f*)(m + b * H_))[1];
  v4f zv0 = ((const v4f*)(zinv + b * H_))[0], zv1 = ((const v4f*)(zinv + b * H_))[1];
  v4f acc[H_] = {};
  for (int base = s0; base < s1; base += 64) {
    int cnt = min(64, s1 - base);
    for (int i4 = t; i4 < cnt * 2; i4 += 256) {
      int n = base + (i4 >> 1);
      v4f lv = ((const v4f*)logit)[(size_t)n * 2 + (i4 & 1)];
      v4f a = (i4 & 1) ? exp4(lv - mh1) * zv1 : exp4(lv - mh0) * zv0;
      attn4[i4] = a;
    }
    __syncthreads();
    for (int k = g; k < cnt; k += 4) {
      int n = base + k;
      v4f x = h4[(size_t)n * 64 + c4];
      v4f hlv = (x - mu_n[n]) * rstd_n[n] * w + bi;
#pragma unroll
      for (int hh = 0; hh < H_; ++hh) acc[hh] += attnf[k * H_ + hh] * hlv;
    }
    __syncthreads();
  }
  // combine the 4 node-subgroups and store
#pragma unroll
  for (int hh = 0; hh < H_; ++hh) {
    red[t] = acc[hh];
    __syncthreads();
    if (g == 0) {
      v4f v = (red[t] + red[t + 64]) + (red[t + 128] + red[t + 192]);
      ((v4f*)s)[((size_t)b * H_ + hh) * 64 + c4] = v;
    }
    __syncthreads();
  }
}

// ---------------------------------------------------------------- K6: g[b,o] = s[b, o>>5, :] . Wv[o,:]
__global__ void g_compute_k(const float* __restrict__ s, const float* __restrict__ Wv,
                            float* __restrict__ g) {
  int b = blockIdx.x, o = threadIdx.x;
  int hh = o >> 5;
  const v4f* srow = (const v4f*)(s + ((size_t)b * H_ + hh) * D_);
  const v4f* wrow = (const v4f*)(Wv + (size_t)o * D_);
  v4f acc = splat4(0.f);
#pragma unroll 4
  for (int c = 0; c < D_ / 4; ++c) acc += srow[c] * wrow[c];
  g[(size_t)b * D_ + o] = hsum4(acc);
}

// ---------------------------------------------------------------- K7: out = g@Wout^T + 0.2*mean@Wres^T (f32 WMMA)
__global__ void out_gemm_k(const float* __restrict__ g, const float* __restrict__ meanr,
                           const float* __restrict__ Wout, const float* __restrict__ Wres,
                           float* __restrict__ out) {
  int wave = (blockIdx.x * blockDim.x + threadIdx.x) >> 5;
  int lane = threadIdx.x & 31;
  int mt = wave >> 4;
  int nt = wave & 15;
  int r  = lane & 15;
  int kh = (lane >> 4) << 1;
  v8f acc = {};
  int ar = (mt * 16 + r) * D_;
  int br = (nt * 16 + r) * D_;
#pragma unroll 4
  for (int kk = 0; kk < D_; kk += 4) {
    int k0 = kk + kh;
    v2f a, bm;
    a.x = g[ar + k0]; a.y = g[ar + k0 + 1];
    bm.x = Wout[br + k0]; bm.y = Wout[br + k0 + 1];
    acc = __builtin_amdgcn_wmma_f32_16x16x4_f32(false, a, false, bm, (short)0, acc,
                                                false, false);
  }
#pragma unroll 4
  for (int kk = 0; kk < D_; kk += 4) {
    int k0 = kk + kh;
    v2f a, bm;
    a.x = MEAN_RES_SCALE_ * meanr[ar + k0]; a.y = MEAN_RES_SCALE_ * meanr[ar + k0 + 1];
    bm.x = Wres[br + k0]; bm.y = Wres[br + k0 + 1];
    acc = __builtin_amdgcn_wmma_f32_16x16x4_f32(false, a, false, bm, (short)0, acc,
                                                false, false);
  }
  int mb  = mt * 16 + ((lane >> 4) << 3);
  int col = nt * 16 + r;
#pragma unroll
  for (int i = 0; i < 8; ++i) out[(mb + i) * D_ + col] = acc[i];
}

// ----------------------------------------------------------------
extern "C" void kernel_launch(void* const* d_in, const int* in_sizes, int n_in,
                              void* d_out, int out_size, void* d_ws, size_t ws_size,
                              hipStream_t stream) {
  (void)in_sizes; (void)n_in; (void)out_size; (void)ws_size;
  const float*   h       = (const float*)d_in[0];
  const int*     batch   = (const int*)d_in[1];
  const uint8_t* cdr     = (const uint8_t*)d_in[2];
  const uint8_t* ifc     = (const uint8_t*)d_in[3];
  const float*   ln_kv_w = (const float*)d_in[4];
  const float*   ln_kv_b = (const float*)d_in[5];
  const float*   ln_q_w  = (const float*)d_in[6];
  const float*   ln_q_b  = (const float*)d_in[7];
  const float*   Wk      = (const float*)d_in[8];
  const float*   Wv      = (const float*)d_in[9];
  const float*   Wq      = (const float*)d_in[10];
  const float*   Wres    = (const float*)d_in[11];
  const float*   Wout    = (const float*)d_in[12];
  const float*   cdr_b   = (const float*)d_in[13];
  const float*   if_b    = (const float*)d_in[14];
  const float*   lscale  = (const float*)d_in[15];
  float* out = (float*)d_out;

  char* w = (char*)d_ws;
  size_t off = 0;
  auto take = [&](size_t bytes) -> void* {
    void* p = w + off;
    off += (bytes + 255) & ~(size_t)255;
    return p;
  };
  int*   seg    = (int*)take((B_ + 1) * sizeof(int));
  float* mu_n   = (float*)take((size_t)N_ * sizeof(float));
  float* rstd_n = (float*)take((size_t)N_ * sizeof(float));
  float* meanr  = (float*)take((size_t)B_ * D_ * sizeof(float));
  float* maxr   = (float*)take((size_t)B_ * D_ * sizeof(float));
  float* qpre   = (float*)take((size_t)B_ * D_ * sizeof(float));
  float* q      = (float*)take((size_t)B_ * D_ * sizeof(float));
  float* P      = (float*)take((size_t)B_ * H_ * D_ * sizeof(float));
  float* logit  = (float*)take((size_t)N_ * H_ * sizeof(float));
  float* m      = (float*)take((size_t)B_ * H_ * sizeof(float));
  float* zinv   = (float*)take((size_t)B_ * H_ * sizeof(float));
  float* s      = (float*)take((size_t)B_ * H_ * D_ * sizeof(float));
  float* g      = (float*)take((size_t)B_ * D_ * sizeof(float));

  seg_offsets_k<<<N_ / 256, 256, 0, stream>>>(batch, seg);
  pool_k<<<B_, 256, 0, stream>>>(h, seg, meanr, maxr);
  q_gemm_k<<<256, 128, 0, stream>>>(meanr, maxr, Wq, qpre);   // 1024 wmma tiles
  q_ln_k<<<B_, D_, 0, stream>>>(qpre, ln_q_w, ln_q_b, q);
  p_compute_k<<<B_ * H_, D_, 0, stream>>>(q, Wk, P);
  logits_k<<<N_ / 8, 256, 0, stream>>>(h, batch, cdr, ifc, ln_kv_w, ln_kv_b, P,
                                       cdr_b, if_b, lscale, logit, mu_n, rstd_n);
  stats_k<<<B_, 256, 0, stream>>>(logit, seg, m, zinv);
  s_accum_k<<<B_, 256, 0, stream>>>(h, seg, logit, m, zinv, mu_n, rstd_n,
                                    ln_kv_w, ln_kv_b, s);
  g_compute_k<<<B_, D_, 0, stream>>>(s, Wv, g);
  out_gemm_k<<<256, 128, 0, stream>>>(g, meanr, Wout, Wres, out);
}